// Eq2Net_7859790151696
// MI455X (gfx1250) — compile-verified
//
#include <hip/hip_runtime.h>

typedef __attribute__((ext_vector_type(2))) float v2f;
typedef __attribute__((ext_vector_type(8))) float v8f;

#define T_STEPS 2048
#define S_DIM   512
#define B_DIM   16
#define A_DIM   18
#define PEN_F   0.5f

// ---------------------------------------------------------------------------
// FP32 WMMA GEMM: C(MxN) = A(MxK) * B(KxN), one 16x16 tile per wave,
// V_WMMA_F32_16X16X4_F32 over K in steps of 4.
//
// A fragment (16x4): lane L holds row M=L%16, VGPR v holds K = 2*(L>>4)+v.
// B fragment (4x16): lane L holds col N=L%16, VGPR v holds K = v + 2*(L>>4).
// C/D (16x16):       lane L holds col N=L%16, VGPR r holds row M = r + 8*(L>>4).
// ---------------------------------------------------------------------------
__global__ void gemm_wmma_f32_16x16x4(const float* __restrict__ A,
                                      const float* __restrict__ B,
                                      float* __restrict__ C,
                                      int M, int N, int K, int tilesN) {
  int wave   = blockIdx.x * (blockDim.x >> 5) + (threadIdx.x >> 5);
  int tilesM = (M + 15) >> 4;
  if (wave >= tilesM * tilesN) return;
  int mt   = wave / tilesN;
  int nt   = wave % tilesN;
  int lane = threadIdx.x & 31;
  int l16  = lane & 15;
  int kgrp = lane >> 4;               // 0 or 1

  int mrow = mt * 16 + l16;
  int aRow = mrow < M ? mrow : (M - 1);   // clamp; partial-tile stores guarded
  int ncol = nt * 16 + l16;

  const float* Ap = A + (size_t)aRow * K + 2 * kgrp;   // + k0 + v
  const float* Bp = B + (size_t)(2 * kgrp) * N + ncol; // + (k0+v)*N

  v8f acc = {};
#pragma unroll 4
  for (int k0 = 0; k0 < K; k0 += 4) {
    v2f a, b;
    a.x = Ap[k0 + 0];
    a.y = Ap[k0 + 1];
    b.x = Bp[(size_t)(k0 + 0) * N];
    b.y = Bp[(size_t)(k0 + 1) * N];
    acc = __builtin_amdgcn_wmma_f32_16x16x4_f32(
        /*neg_a=*/false, a, /*neg_b=*/false, b,
        /*c_mod=*/(short)0, acc, /*reuse_a=*/false, /*reuse_b=*/false);
  }

#pragma unroll
  for (int r = 0; r < 8; ++r) {
    int m = mt * 16 + r + 8 * kgrp;
    if (m < M) C[(size_t)m * N + ncol] = acc[r];
  }
}

// ---------------------------------------------------------------------------
// Per-(t,b): gather chosen action logit and subtract logsumexp over A=18.
// ---------------------------------------------------------------------------
__global__ void action_gather_lse(const float* __restrict__ logits,
                                  const int* __restrict__ actions,
                                  float* __restrict__ aLp) {
  int idx = blockIdx.x * blockDim.x + threadIdx.x;
  if (idx >= T_STEPS * B_DIM) return;
  int i = idx >> 4, b = idx & 15;
  const float* row = logits + (size_t)i * (B_DIM * A_DIM) + b * A_DIM;
  float m = row[0];
#pragma unroll
  for (int a = 1; a < A_DIM; ++a) m = fmaxf(m, row[a]);
  float s = 0.0f;
#pragma unroll
  for (int a = 0; a < A_DIM; ++a) s += expf(row[a] - m);
  int act = actions[i];
  aLp[idx] = row[act] - (m + logf(s));
}

// ---------------------------------------------------------------------------
// Per-(t,b): 2-way log_softmax of stop logits -> stop0n, stop1n.
// ---------------------------------------------------------------------------
__global__ void stop_norm(const float* __restrict__ logits,
                          float* __restrict__ s0n, float* __restrict__ s1n) {
  int idx = blockIdx.x * blockDim.x + threadIdx.x;
  if (idx >= (T_STEPS + 1) * B_DIM) return;
  int t = idx >> 4, b = idx & 15;
  float x0 = logits[(size_t)t * (2 * B_DIM) + 2 * b + 0];
  float x1 = logits[(size_t)t * (2 * B_DIM) + 2 * b + 1];
  float m = fmaxf(x0, x1);
  float l = m + logf(expf(x0 - m) + expf(x1 - m));
  s0n[idx] = x0 - l;
  s1n[idx] = x1 - l;
}

// ---------------------------------------------------------------------------
// Per-t: 16-way log_softmax of start logits.
// ---------------------------------------------------------------------------
__global__ void start_norm(const float* __restrict__ logits,
                           float* __restrict__ stn) {
  int t = blockIdx.x * blockDim.x + threadIdx.x;
  if (t >= T_STEPS + 1) return;
  const float* row = logits + (size_t)t * B_DIM;
  float m = row[0];
#pragma unroll
  for (int b = 1; b < B_DIM; ++b) m = fmaxf(m, row[b]);
  float s = 0.0f;
#pragma unroll
  for (int b = 0; b < B_DIM; ++b) s += expf(row[b] - m);
  float l = m + logf(s);
#pragma unroll
  for (int b = 0; b < B_DIM; ++b) stn[(size_t)t * B_DIM + b] = row[b] - l;
}

// ---------------------------------------------------------------------------
// Wave-wide logsumexp over 32 lanes (inactive lanes carry -inf -> contribute 0)
// ---------------------------------------------------------------------------
__device__ __forceinline__ float lse_wave(float x) {
  float m = x;
#pragma unroll
  for (int off = 16; off; off >>= 1) m = fmaxf(m, __shfl_xor(m, off, 32));
  float e = expf(x - m);          // x = -inf, m finite -> 0
#pragma unroll
  for (int off = 16; off; off >>= 1) e += __shfl_xor(e, off, 32);
  return m + logf(e);
}

// ---------------------------------------------------------------------------
// Serial HMM scan, O(T*B) via per-column running logsumexp:
//   col[b] <- logaddexp(col[b]+stop0_i[b], start_i[b] + lse_b(col+stop1_i) - PEN)
// One wave; lanes 0..15 carry the B columns.
// ---------------------------------------------------------------------------
__global__ void hmm_scan(const float* __restrict__ aLp,
                         const float* __restrict__ s0n,
                         const float* __restrict__ s1n,
                         const float* __restrict__ stn,
                         float* __restrict__ out) {
  const int lane = threadIdx.x;
  const bool on  = lane < B_DIM;
  const float NINF = -__builtin_inff();

  float col = on ? stn[lane] : NINF;             // start_logps[0]
  float a0  = on ? aLp[lane] : 0.0f;
  float tot = lse_wave(on ? (a0 + col) : NINF);  // step i = 0 (buffer unchanged)

  for (int i = 1; i < T_STEPS; ++i) {
    int base = i * B_DIM + (on ? lane : 0);
    float s1 = s1n[base];
    float s0 = s0n[base];
    float st = stn[base];
    float a  = aLp[base];
    int pf = (i + 8 < T_STEPS) ? (i + 8) : (T_STEPS - 1);
    __builtin_prefetch(&aLp[pf * B_DIM + lane], 0, 0);
    __builtin_prefetch(&s0n[pf * B_DIM + lane], 0, 0);

    float tr = lse_wave(on ? (col + s1) : NINF) - PEN_F;
    float nm = st + tr;                 // new_mass
    float x  = col + s0;
    float m  = fmaxf(x, nm);
    float nc = m + logf(expf(x - m) + expf(nm - m));
    col = on ? nc : NINF;
    tot += lse_wave(on ? (a + col) : NINF);
  }

  float f = on ? s0n[T_STEPS * B_DIM + lane] : 0.0f;   // stop_logps[-1][:,0]
  tot += lse_wave(on ? (col + f) : NINF);

  if (lane == 0) out[0] = tot;
}

// ---------------------------------------------------------------------------
extern "C" void kernel_launch(void* const* d_in, const int* in_sizes, int n_in,
                              void* d_out, int out_size, void* d_ws, size_t ws_size,
                              hipStream_t stream) {
  const float* s_i     = (const float*)d_in[0];   // (T+1, S)
  const float* W_act   = (const float*)d_in[1];   // (S, B*A)
  const float* W_stop  = (const float*)d_in[2];   // (S, B*2)
  const float* W_start = (const float*)d_in[3];   // (S, B)
  const int*   actions = (const int*)d_in[4];     // (T,)

  const int Tp1 = T_STEPS + 1;
  float* ws  = (float*)d_ws;
  float* lgA = ws;                                // Tp1 * 288
  float* lgS = lgA + (size_t)Tp1 * (B_DIM * A_DIM);
  float* lgT = lgS + (size_t)Tp1 * (B_DIM * 2);   // Tp1 * 16
  float* aLp = lgT + (size_t)Tp1 * B_DIM;         // T   * 16
  float* s0n = aLp + (size_t)T_STEPS * B_DIM;     // Tp1 * 16
  float* s1n = s0n + (size_t)Tp1 * B_DIM;         // Tp1 * 16
  float* stn = s1n + (size_t)Tp1 * B_DIM;         // Tp1 * 16

  const int wavesPerBlock = 8;                    // 256 threads = 8 wave32
  auto gemm = [&](const float* Bw, float* C, int N) {
    int tilesM = (Tp1 + 15) / 16;
    int tilesN = N / 16;
    int waves  = tilesM * tilesN;
    int grid   = (waves + wavesPerBlock - 1) / wavesPerBlock;
    gemm_wmma_f32_16x16x4<<<grid, 256, 0, stream>>>(s_i, Bw, C, Tp1, N, S_DIM, tilesN);
  };
  gemm(W_act,   lgA, B_DIM * A_DIM);  // 2049 x 288
  gemm(W_stop,  lgS, B_DIM * 2);      // 2049 x 32
  gemm(W_start, lgT, B_DIM);          // 2049 x 16

  action_gather_lse<<<(T_STEPS * B_DIM + 255) / 256, 256, 0, stream>>>(lgA, actions, aLp);
  stop_norm<<<(Tp1 * B_DIM + 255) / 256, 256, 0, stream>>>(lgS, s0n, s1n);
  start_norm<<<(Tp1 + 255) / 256, 256, 0, stream>>>(lgT, stn);

  hmm_scan<<<1, 32, 0, stream>>>(aLp, s0n, s1n, stn, (float*)d_out);
}